// GAT_QNetwork_84524956385462
// MI455X (gfx1250) — compile-verified
//
#include <hip/hip_runtime.h>
#include <hip/hip_bf16.h>

// ---------------------------------------------------------------------------
// GAT QNetwork forward: 2x GATConv + FC head, CDNA5 (gfx1250, wave32, WMMA).
//   GEMMs: exact-f32 V_WMMA_F32_16X16X4_F32, weight panel staged in LDS
//   (transposed, padded, conflict-free ds loads), async global->LDS copy
//   (GLOBAL_LOAD_ASYNC_TO_LDS_B32, ASYNCcnt). B fragments batch-loaded into
//   registers so DS latency pipelines across 16 independent WMMAs.
//   Edge phase: 3 passes (max / denom / alpha) + float4 weighted scatter
//   with global_atomic_add_f32 (L2-resident).
// ---------------------------------------------------------------------------

typedef __attribute__((ext_vector_type(2))) float v2f;
typedef __attribute__((ext_vector_type(8))) float v8f;

#define IN_CH 128
#define HIDC  64
#define ACTS  5
#define NEG_SLOPE 0.2f
#define NEG_INF_ORD ((int)0x807FFFFF)   // f2ord(-inf)

__device__ __forceinline__ int   f2ord(float f) { int i = __float_as_int(f); return i >= 0 ? i : (i ^ 0x7fffffff); }
__device__ __forceinline__ float ord2f(int i)   { return __int_as_float(i >= 0 ? i : (i ^ 0x7fffffff)); }
__device__ __forceinline__ float lrelu(float v) { return v > 0.f ? v : NEG_SLOPE * v; }

// ---------------- workspace init (accumulators must be re-zeroed per call) --
__global__ __launch_bounds__(256)
void init_kernel(float* agg1, float* agg2, float* dn1, float* dn2,
                 int* em1, int* em2, int n)
{
    int idx = blockIdx.x * blockDim.x + threadIdx.x;
    if (idx < n * 256) agg1[idx] = 0.f;
    if (idx < n * 64)  agg2[idx] = 0.f;
    if (idx < n * 4)  { dn1[idx] = 0.f; em1[idx] = NEG_INF_ORD; }
    if (idx < n)      { dn2[idx] = 0.f; em2[idx] = NEG_INF_ORD; }
}

// ---------------- dense GEMM C[M,NC] = A[M,K] @ B[K,NC] --------------------
// Block: 256 threads (8 waves). Block tile: 128 rows x 64 cols.
// B panel (K x 64) staged in LDS transposed: lds[col*(K+4) + k].
//   - stride K+4 == 4 (mod 64): half-wave read banks 4r/4r+1 -> conflict-free
//   - k + 2*half even -> 8B-aligned 64-bit ds loads per fragment
// Wave computes 16x64 via 4x V_WMMA_F32_16X16X4_F32 accumulators.
__global__ __launch_bounds__(256)
void gemm_f32_wmma(const float* __restrict__ A, const float* __restrict__ B,
                   float* __restrict__ C, int M, int K, int NC)
{
    extern __shared__ float lds[];                 // 64 * (K+4) floats
    const int tid  = threadIdx.x;
    const int lane = tid & 31;
    const int wave = tid >> 5;
    const int half = lane >> 4;
    const int r    = lane & 15;
    const int n0   = blockIdx.x * 64;
    const int ldb  = K + 4;

    // ---- cooperative transposed panel load: lds[c*ldb + k] = B[k][n0+c] ----
    {
        const int c  = tid & 63;       // column within strip (coalesced dim)
        const int k0 = tid >> 6;       // 0..3
        for (int k = k0; k < K; k += 4) {
#if __has_builtin(__builtin_amdgcn_global_load_async_to_lds_b32)
            __builtin_amdgcn_global_load_async_to_lds_b32(
                (__attribute__((address_space(1))) int*)const_cast<float*>(B + (size_t)k * NC + n0 + c),
                (__attribute__((address_space(3))) int*)(lds + c * ldb + k),
                0, 0);
#else
            lds[c * ldb + k] = B[(size_t)k * NC + n0 + c];
#endif
        }
    }
#if __has_builtin(__builtin_amdgcn_global_load_async_to_lds_b32)
#if __has_builtin(__builtin_amdgcn_s_wait_asynccnt)
    __builtin_amdgcn_s_wait_asynccnt(0);
#else
    asm volatile("s_wait_asynccnt 0x0" ::: "memory");
#endif
#endif
    __syncthreads();

    const int tileM = blockIdx.y * 8 + wave;       // wave-uniform guard
    if (tileM * 16 < M) {
        v8f acc[4] = {};
        const float* arow = A + (size_t)(tileM * 16 + r) * K + 2 * half;
        const float* brow = lds + r * ldb + 2 * half;
        for (int kb = 0; kb < K; kb += 16) {
            float2 af[4];
#pragma unroll
            for (int u = 0; u < 4; ++u)            // batch A loads (global)
                af[u] = *(const float2*)(arow + kb + 4 * u);
            float2 bf[4][4];
#pragma unroll
            for (int u = 0; u < 4; ++u)            // batch 16 B loads (LDS):
#pragma unroll
                for (int t = 0; t < 4; ++t)        // 16 outstanding ds ops
                    bf[u][t] = *(const float2*)(brow + (t * 16) * ldb + kb + 4 * u);
#pragma unroll
            for (int u = 0; u < 4; ++u) {
                v2f a; a.x = af[u].x; a.y = af[u].y;
#pragma unroll
                for (int t = 0; t < 4; ++t) {
                    v2f b; b.x = bf[u][t].x; b.y = bf[u][t].y;
                    acc[t] = __builtin_amdgcn_wmma_f32_16x16x4_f32(
                        false, a, false, b, (short)0, acc[t], false, false);
                }
            }
        }
#pragma unroll
        for (int t = 0; t < 4; ++t) {
#pragma unroll
            for (int i = 0; i < 8; ++i) {
                // C/D layout: VGPR i -> row i + 8*half, col = lane%16
                C[(size_t)(tileM * 16 + i + 8 * half) * NC + n0 + t * 16 + r] = acc[t][i];
            }
        }
    }
}

// ---------------- per-(node,head) attention coefficients -------------------
__global__ __launch_bounds__(256)
void alpha_kernel(const float* __restrict__ h, const float* __restrict__ avs,
                  const float* __restrict__ avd, float* __restrict__ os,
                  float* __restrict__ od, int n, int heads)
{
    int idx = blockIdx.x * blockDim.x + threadIdx.x;
    if (idx >= n * heads) return;
    int node = idx / heads, hd = idx % heads;
    const float* hp = h + (size_t)node * heads * HIDC + hd * HIDC;
    const float* s  = avs + hd * HIDC;
    const float* d  = avd + hd * HIDC;
    float accs = 0.f, accd = 0.f;
#pragma unroll
    for (int c = 0; c < HIDC; ++c) { float v = hp[c]; accs += v * s[c]; accd += v * d[c]; }
    os[idx] = accs;
    od[idx] = accd;
}

__device__ __forceinline__ void edge_sd(const int* ei, int E, int e, int& s, int& d)
{
    if (e < E) { s = ei[e]; d = ei[E + e]; } else { s = d = e - E; }  // self-loops appended
}

// ---------------- pass 1: segment max of edge logits -----------------------
__global__ __launch_bounds__(256)
void edge_max_kernel(const int* __restrict__ ei, int E, int Et,
                     const float* __restrict__ as, const float* __restrict__ ad,
                     int* __restrict__ emax, int heads)
{
    int idx = blockIdx.x * blockDim.x + threadIdx.x;
    if (idx >= Et * heads) return;
    int e = idx / heads, hd = idx % heads;
    int s, d; edge_sd(ei, E, e, s, d);
    float v = lrelu(as[s * heads + hd] + ad[d * heads + hd]);
    atomicMax(&emax[d * heads + hd], f2ord(v));
}

// ---------------- pass 2: segment sum of exp(logit - max) ------------------
__global__ __launch_bounds__(256)
void edge_denom_kernel(const int* __restrict__ ei, int E, int Et,
                       const float* __restrict__ as, const float* __restrict__ ad,
                       const int* __restrict__ emax, float* __restrict__ denom, int heads)
{
    int idx = blockIdx.x * blockDim.x + threadIdx.x;
    if (idx >= Et * heads) return;
    int e = idx / heads, hd = idx % heads;
    int s, d; edge_sd(ei, E, e, s, d);
    float v = lrelu(as[s * heads + hd] + ad[d * heads + hd]);
    atomicAdd(&denom[d * heads + hd], expf(v - ord2f(emax[d * heads + hd])));
}

// ---------------- pass 3: materialize per-edge alpha (read once in scatter) -
__global__ __launch_bounds__(256)
void edge_alpha_kernel(const int* __restrict__ ei, int E, int Et,
                       const float* __restrict__ as, const float* __restrict__ ad,
                       const int* __restrict__ emax, const float* __restrict__ denom,
                       float* __restrict__ ae, int heads)
{
    int idx = blockIdx.x * blockDim.x + threadIdx.x;
    if (idx >= Et * heads) return;
    int e = idx / heads, hd = idx % heads;
    int s, d; edge_sd(ei, E, e, s, d);
    float v = lrelu(as[s * heads + hd] + ad[d * heads + hd]);
    ae[idx] = expf(v - ord2f(emax[d * heads + hd])) / (denom[d * heads + hd] + 1e-16f);
}

// ---------------- alpha-weighted feature scatter (float4 per thread) -------
__global__ __launch_bounds__(256)
void aggregate_kernel(const int* __restrict__ ei, int E, int Et,
                      const float* __restrict__ h, const float* __restrict__ ae,
                      float* __restrict__ agg, int heads)
{
    const int gpe = heads * (HIDC / 4);                 // 4-channel groups per edge
    long long idx = (long long)blockIdx.x * blockDim.x + threadIdx.x;
    if (idx >= (long long)Et * gpe) return;
    int e  = (int)(idx / gpe);
    int c0 = (int)(idx % gpe) * 4;
    int hd = c0 >> 6;                                   // HIDC == 64
    int s, d; edge_sd(ei, E, e, s, d);
    float alpha = ae[e * heads + hd];
    const float4 hv = *(const float4*)(h + (size_t)s * heads * HIDC + c0);
    float* o = agg + (size_t)d * heads * HIDC + c0;
    atomicAdd(o + 0, hv.x * alpha);
    atomicAdd(o + 1, hv.y * alpha);
    atomicAdd(o + 2, hv.z * alpha);
    atomicAdd(o + 3, hv.w * alpha);
}

// ---------------- bias + ReLU (in place) -----------------------------------
__global__ __launch_bounds__(256)
void bias_relu_kernel(float* __restrict__ buf, const float* __restrict__ bias,
                      long long total, int C)
{
    long long idx = (long long)blockIdx.x * blockDim.x + threadIdx.x;
    if (idx >= total) return;
    float v = buf[idx] + bias[idx % C];
    buf[idx] = v > 0.f ? v : 0.f;
}

// ---------------- FC head: out[n,5] = h[n,64] @ Wfc + bfc -------------------
__global__ __launch_bounds__(256)
void fc_kernel(const float* __restrict__ h, const float* __restrict__ W,
               const float* __restrict__ bias, float* __restrict__ out, int n)
{
    int idx = blockIdx.x * blockDim.x + threadIdx.x;
    if (idx >= n * ACTS) return;
    int node = idx / ACTS, a = idx % ACTS;
    const float* hp = h + (size_t)node * HIDC;
    float acc = bias[a];
#pragma unroll
    for (int k = 0; k < HIDC; ++k) acc += hp[k] * W[k * ACTS + a];
    out[idx] = acc;
}

static inline unsigned cdiv(long long a, long long b) { return (unsigned)((a + b - 1) / b); }

extern "C" void kernel_launch(void* const* d_in, const int* in_sizes, int n_in,
                              void* d_out, int out_size, void* d_ws, size_t ws_size,
                              hipStream_t stream)
{
    const float* x   = (const float*)d_in[0];
    const int*   ei  = (const int*)  d_in[1];
    const float* W1  = (const float*)d_in[2];
    const float* a1s = (const float*)d_in[3];
    const float* a1d = (const float*)d_in[4];
    const float* b1  = (const float*)d_in[5];
    const float* W2  = (const float*)d_in[6];
    const float* a2s = (const float*)d_in[7];
    const float* a2d = (const float*)d_in[8];
    const float* b2  = (const float*)d_in[9];
    const float* Wfc = (const float*)d_in[10];
    const float* bfc = (const float*)d_in[11];
    float* out = (float*)d_out;

    const int n  = in_sizes[0] / IN_CH;   // 50000
    const int E  = in_sizes[1] / 2;       // 800000
    const int Et = E + n;                 // + self loops

    // workspace carve-up (floats): ~150 MB total
    float* h1   = (float*)d_ws;            // n*256
    float* agg1 = h1   + (size_t)n * 256;  // n*256
    float* h2   = agg1 + (size_t)n * 256;  // n*64
    float* agg2 = h2   + (size_t)n * 64;   // n*64
    float* as1  = agg2 + (size_t)n * 64;   // n*4
    float* ad1  = as1  + (size_t)n * 4;    // n*4
    float* dn1  = ad1  + (size_t)n * 4;    // n*4
    int*   em1  = (int*)(dn1 + (size_t)n * 4);   // n*4
    float* as2  = (float*)(em1 + (size_t)n * 4); // n
    float* ad2  = as2 + n;                 // n
    float* dn2  = ad2 + n;                 // n
    int*   em2  = (int*)(dn2 + n);         // n
    float* ae1  = (float*)(em2 + n);       // Et*4
    float* ae2  = ae1 + (size_t)Et * 4;    // Et

    const int TB = 256;

    // 0) reset per-call accumulators
    init_kernel<<<cdiv((long long)n * 256, TB), TB, 0, stream>>>(agg1, agg2, dn1, dn2, em1, em2, n);

    // ---- layer 1 (4 heads x 64) ----
    gemm_f32_wmma<<<dim3(256 / 64, cdiv(n, 128)), 256, (size_t)64 * (IN_CH + 4) * 4, stream>>>(
        x, W1, h1, n, IN_CH, 256);
    alpha_kernel<<<cdiv((long long)n * 4, TB), TB, 0, stream>>>(h1, a1s, a1d, as1, ad1, n, 4);
    edge_max_kernel<<<cdiv((long long)Et * 4, TB), TB, 0, stream>>>(ei, E, Et, as1, ad1, em1, 4);
    edge_denom_kernel<<<cdiv((long long)Et * 4, TB), TB, 0, stream>>>(ei, E, Et, as1, ad1, em1, dn1, 4);
    edge_alpha_kernel<<<cdiv((long long)Et * 4, TB), TB, 0, stream>>>(ei, E, Et, as1, ad1, em1, dn1, ae1, 4);
    aggregate_kernel<<<cdiv((long long)Et * 64, TB), TB, 0, stream>>>(ei, E, Et, h1, ae1, agg1, 4);
    bias_relu_kernel<<<cdiv((long long)n * 256, TB), TB, 0, stream>>>(agg1, b1, (long long)n * 256, 256);

    // ---- layer 2 (1 head x 64) ----
    gemm_f32_wmma<<<dim3(64 / 64, cdiv(n, 128)), 256, (size_t)64 * (256 + 4) * 4, stream>>>(
        agg1, W2, h2, n, 256, 64);
    alpha_kernel<<<cdiv((long long)n, TB), TB, 0, stream>>>(h2, a2s, a2d, as2, ad2, n, 1);
    edge_max_kernel<<<cdiv((long long)Et, TB), TB, 0, stream>>>(ei, E, Et, as2, ad2, em2, 1);
    edge_denom_kernel<<<cdiv((long long)Et, TB), TB, 0, stream>>>(ei, E, Et, as2, ad2, em2, dn2, 1);
    edge_alpha_kernel<<<cdiv((long long)Et, TB), TB, 0, stream>>>(ei, E, Et, as2, ad2, em2, dn2, ae2, 1);
    aggregate_kernel<<<cdiv((long long)Et * 16, TB), TB, 0, stream>>>(ei, E, Et, h2, ae2, agg2, 1);
    bias_relu_kernel<<<cdiv((long long)n * 64, TB), TB, 0, stream>>>(agg2, b2, (long long)n * 64, 64);

    // ---- FC head ----
    fc_kernel<<<cdiv((long long)n * ACTS, TB), TB, 0, stream>>>(agg2, Wfc, bfc, out, n);
}